// YOLOLoss_43568148251316
// MI455X (gfx1250) — compile-verified
//
#include <hip/hip_runtime.h>
#include <math.h>
#include <stdint.h>

// ---------------------------------------------------------------------------
// YOLO loss for MI455X (gfx1250). Key insight: only 3/255 channels (pobj) are
// dense; box/cls terms are sparse gathers at <=512 target cells per level.
// Dense obj path streams through LDS via gfx1250 async global->LDS ops.
// ---------------------------------------------------------------------------

#define B_ 16
#define T_ 32
#define NCLS 80

__constant__ float c_anch[3][3][2] = {
    {{10.f, 13.f}, {16.f, 30.f}, {33.f, 23.f}},
    {{30.f, 61.f}, {62.f, 45.f}, {59.f, 119.f}},
    {{116.f, 90.f}, {156.f, 198.f}, {373.f, 326.f}}};
__constant__ float c_strd[3] = {8.f, 16.f, 32.f};
__constant__ int   c_dim[3]  = {80, 40, 20};
__constant__ int   c_base[3] = {0, 307200, 384000};  // cell-array base per level

// total cells = 16*3*(6400+1600+400) = 403200
#define CELLS_TOTAL 403200
#define WS_ACC_OFF   0      // 10 floats: [0]=box [1..3]=obj_sum [4..6]=cls_sum [7..9]=n
#define WS_OWNER_OFF 64     // 1536 ints (cell+1 if owner else 0)
#define WS_CELLS_OFF 8192   // CELLS_TOTAL x uint4 {flag, clsmask0, clsmask1, clsmask2}
#define WS_TOTAL_U4  ((WS_CELLS_OFF + CELLS_TOTAL * 16) / 16)

__device__ __forceinline__ float bce_logits(float x, float t) {
    return fmaxf(x, 0.f) - x * t + log1pf(expf(-fabsf(x)));
}
__device__ __forceinline__ float sigm(float x) { return 1.f / (1.f + expf(-x)); }

__device__ float ciou_val(float b1x, float b1y, float b1w, float b1h,
                          float b2x, float b2y, float b2w, float b2h) {
    float b1x1 = b1x - b1w * 0.5f, b1y1 = b1y - b1h * 0.5f;
    float b1x2 = b1x + b1w * 0.5f, b1y2 = b1y + b1h * 0.5f;
    float b2x1 = b2x - b2w * 0.5f, b2y1 = b2y - b2h * 0.5f;
    float b2x2 = b2x + b2w * 0.5f, b2y2 = b2y + b2h * 0.5f;
    float iw = fmaxf(fminf(b1x2, b2x2) - fmaxf(b1x1, b2x1), 0.f);
    float ih = fmaxf(fminf(b1y2, b2y2) - fmaxf(b1y1, b2y1), 0.f);
    float inter = iw * ih;
    float a1 = (b1x2 - b1x1) * (b1y2 - b1y1);
    float a2 = (b2x2 - b2x1) * (b2y2 - b2y1);
    float iou = inter / (a1 + a2 - inter + 1e-6f);
    float ex1 = fminf(b1x1, b2x1), ey1 = fminf(b1y1, b2y1);
    float ex2 = fmaxf(b1x2, b2x2), ey2 = fmaxf(b1y2, b2y2);
    float cdist = (b1x - b2x) * (b1x - b2x) + (b1y - b2y) * (b1y - b2y);
    float diag  = (ex2 - ex1) * (ex2 - ex1) + (ey2 - ey1) * (ey2 - ey1);
    float diou = iou - cdist / (diag + 1e-6f);
    float dv = atanf(b2w / b2h) - atanf(b1w / b1h);
    float v = (4.0f / (float)(M_PI * M_PI)) * dv * dv;
    float alpha = v / (1.0f - diou + v + 1e-6f);
    return diou - alpha * v;
}

// ---------------------------------------------------------------------------
__global__ __launch_bounds__(256) void init_ws_kernel(uint4* ws, int n4) {
    int i = blockIdx.x * blockDim.x + threadIdx.x;
    int stride = gridDim.x * blockDim.x;
    uint4 z; z.x = 0u; z.y = 0u; z.z = 0u; z.w = 0u;
    for (; i < n4; i += stride) ws[i] = z;
}

// ---------------------------------------------------------------------------
// One thread per (level, b, t): box loss + scatter tobj flag / class bits.
__global__ __launch_bounds__(256) void targets_kernel(
    const float* __restrict__ boxes, const int* __restrict__ labels,
    const float* __restrict__ p0, const float* __restrict__ p1,
    const float* __restrict__ p2, float* __restrict__ acc,
    int* __restrict__ owner, uint4* __restrict__ cells) {
    int idx = blockIdx.x * blockDim.x + threadIdx.x;
    if (idx >= 3 * B_ * T_) return;
    int l = idx / (B_ * T_);
    int r = idx % (B_ * T_);
    int b = r / T_, t = r % T_;

    const float* bx = boxes + (size_t)(b * T_ + t) * 4;
    float x1 = bx[0], y1 = bx[1], x2 = bx[2], y2 = bx[3];
    float cx = (x1 + x2) * 0.5f, cy = (y1 + y2) * 0.5f;
    float w = x2 - x1, h = y2 - y1;
    bool valid = (w > 0.f) && (h > 0.f);

    float s = c_strd[l];
    int W = c_dim[l], H = W, HW = W * H;
    float fx = floorf(cx / s), fy = floorf(cy / s);
    bool in_grid = (fx >= 0.f) && (fx < (float)W) && (fy >= 0.f) && (fy < (float)H);
    if (!(valid && in_grid)) return;  // owner[] already zeroed
    int gx = (int)fx, gy = (int)fy;

    // best anchor: argmin over max-ratio (strict < keeps first on ties = argmin)
    int best = 0; float bv = 3.4e38f;
    #pragma unroll
    for (int a = 0; a < 3; ++a) {
        float aw = c_anch[l][a][0], ah = c_anch[l][a][1];
        float rw = w / aw; rw = fmaxf(rw, 1.f / rw);
        float rh = h / ah; rh = fmaxf(rh, 1.f / rh);
        float m = fmaxf(rw, rh);
        if (m < bv) { bv = m; best = a; }
    }

    int cell = c_base[l] + ((b * 3 + best) * H + gy) * W + gx;
    unsigned int* cw = (unsigned int*)&cells[cell];
    int lbl = labels[b * T_ + t];
    if (lbl < NCLS) {  // cls_v = vmask & (label < C); lblc = clip(label,0,C-1)
        int lc = lbl < 0 ? 0 : lbl;
        atomicOr(&cw[1 + (lc >> 5)], 1u << (lc & 31));
    }
    unsigned int old = atomicCAS(&cw[0], 0u, 1u);
    owner[idx] = (old == 0u) ? (cell + 1) : 0;

    // box loss: gather 4 logits of chosen anchor at target cell
    const float* pred = (l == 0) ? p0 : ((l == 1) ? p1 : p2);
    size_t c0 = ((size_t)(b * 255 + best * 85)) * HW + (size_t)gy * W + gx;
    float q0 = pred[c0], q1 = pred[c0 + (size_t)HW];
    float q2 = pred[c0 + 2 * (size_t)HW], q3 = pred[c0 + 3 * (size_t)HW];
    float aw = c_anch[l][best][0], ah = c_anch[l][best][1];
    float pbx = (sigm(q0) + (float)gx) * s;
    float pby = (sigm(q1) + (float)gy) * s;
    float pbw = expf(q2) * aw * s;
    float pbh = expf(q3) * ah * s;
    float term = 1.0f - ciou_val(pbx, pby, pbw, pbh, cx, cy, w, h);
    atomicAdd(&acc[0], term);
}

// ---------------------------------------------------------------------------
// Owners (one per unique masked cell) sum bce over 80 classes vs bitmask.
__global__ __launch_bounds__(256) void cls_kernel(
    const float* __restrict__ p0, const float* __restrict__ p1,
    const float* __restrict__ p2, float* __restrict__ acc,
    const int* __restrict__ owner, const uint4* __restrict__ cells) {
    int idx = blockIdx.x * blockDim.x + threadIdx.x;
    if (idx >= 3 * B_ * T_) return;
    int enc = owner[idx];
    if (!enc) return;
    int l = idx / (B_ * T_);
    int cell = enc - 1;
    int W = c_dim[l], H = W, HW = W * H;
    int rel = cell - c_base[l];
    int gx = rel % W; int tmp = rel / W;
    int gy = tmp % H; int plane = tmp / H;
    int b = plane / 3, a = plane % 3;
    uint4 m4 = cells[cell];
    unsigned int m[3] = {m4.y, m4.z, m4.w};
    const float* pred = (l == 0) ? p0 : ((l == 1) ? p1 : p2);
    size_t base5 = ((size_t)(b * 255 + a * 85 + 5)) * HW + (size_t)gy * W + gx;
    float sum = 0.f;
    for (int c = 0; c < NCLS; ++c) {
        float x = pred[base5 + (size_t)c * HW];
        float tv = ((m[c >> 5] >> (c & 31)) & 1u) ? 1.f : 0.f;
        sum += bce_logits(x, tv);
    }
    atomicAdd(&acc[4 + l], sum);
    atomicAdd(&acc[7 + l], 1.0f);  // n = #cells with tobj>0
}

// ---------------------------------------------------------------------------
// Dense obj loss over one level. Streams pobj (channel a*85+4) through LDS
// using gfx1250 async global->LDS b128 loads (ASYNCcnt path).
__global__ __launch_bounds__(256) void obj_kernel(
    const float* __restrict__ pred, const unsigned int* __restrict__ cellwords,
    float* __restrict__ acc, int HW, int base, int acc_idx) {
    __shared__ float tile[1024];
    __shared__ float red[256];
    int tid = threadIdx.x;
    int plane = blockIdx.y;             // b*3 + a
    int b = plane / 3, a = plane % 3;
    int pix0 = blockIdx.x * 1024 + tid * 4;
    const float* src = pred + ((size_t)(b * 255 + a * 85 + 4)) * HW;

    if (pix0 < HW) {
        // LDS byte address == low 32 bits of the generic address of a shared var
        uint32_t laddr = (uint32_t)(uintptr_t)(&tile[tid * 4]);
        uint64_t gaddr = (uint64_t)(uintptr_t)(src + pix0);
        asm volatile("global_load_async_to_lds_b128 %0, %1, off"
                     :: "v"(laddr), "v"(gaddr) : "memory");
    }
    asm volatile("s_wait_asynccnt 0" ::: "memory");

    float local = 0.f;
    if (pix0 < HW) {
        const unsigned int* flags = cellwords + (size_t)(base + plane * HW) * 4;
        #pragma unroll
        for (int j = 0; j < 4; ++j) {
            int pix = pix0 + j;
            float x = tile[tid * 4 + j];
            float tv = (flags[(size_t)pix * 4] != 0u) ? 1.f : 0.f;
            local += bce_logits(x, tv);
        }
    }
    red[tid] = local;
    __syncthreads();
    for (int s2 = 128; s2 > 0; s2 >>= 1) {
        if (tid < s2) red[tid] += red[tid + s2];
        __syncthreads();
    }
    if (tid == 0) atomicAdd(&acc[acc_idx], red[0]);
}

// ---------------------------------------------------------------------------
__global__ void finalize_kernel(const float* __restrict__ acc, float* __restrict__ out) {
    if (threadIdx.x == 0 && blockIdx.x == 0) {
        float obj_l = acc[1] / 307200.f + acc[2] / 76800.f + acc[3] / 19200.f;
        float cls_l = 0.f;
        #pragma unroll
        for (int l = 0; l < 3; ++l) {
            float n = acc[7 + l];
            if (n > 0.f) cls_l += acc[4 + l] / (n * (float)NCLS);
        }
        out[0] = 0.05f * acc[0] + 1.0f * obj_l + 0.5f * cls_l;
    }
}

// ---------------------------------------------------------------------------
extern "C" void kernel_launch(void* const* d_in, const int* in_sizes, int n_in,
                              void* d_out, int out_size, void* d_ws, size_t ws_size,
                              hipStream_t stream) {
    const float* p0    = (const float*)d_in[0];
    const float* p1    = (const float*)d_in[1];
    const float* p2    = (const float*)d_in[2];
    const float* boxes = (const float*)d_in[3];
    const int*   labels = (const int*)d_in[4];
    float* out = (float*)d_out;

    float* acc  = (float*)d_ws;
    int* owner  = (int*)((char*)d_ws + WS_OWNER_OFF);
    uint4* cells = (uint4*)((char*)d_ws + WS_CELLS_OFF);
    const unsigned int* cellwords = (const unsigned int*)cells;

    (void)in_sizes; (void)n_in; (void)out_size; (void)ws_size;

    init_ws_kernel<<<512, 256, 0, stream>>>((uint4*)d_ws, WS_TOTAL_U4);
    targets_kernel<<<6, 256, 0, stream>>>(boxes, labels, p0, p1, p2, acc, owner, cells);
    cls_kernel<<<6, 256, 0, stream>>>(p0, p1, p2, acc, owner, cells);
    obj_kernel<<<dim3(7, 48), 256, 0, stream>>>(p0, cellwords, acc, 6400, 0,      1);
    obj_kernel<<<dim3(2, 48), 256, 0, stream>>>(p1, cellwords, acc, 1600, 307200, 2);
    obj_kernel<<<dim3(1, 48), 256, 0, stream>>>(p2, cellwords, acc, 400,  384000, 3);
    finalize_kernel<<<1, 64, 0, stream>>>(acc, out);
}